// GINConv_16131897164158
// MI455X (gfx1250) — compile-verified
//
#include <hip/hip_runtime.h>

typedef __attribute__((ext_vector_type(2))) float v2f;
typedef __attribute__((ext_vector_type(8))) float v8f;

#define N_NODES   100000
#define DEGREE    16
#define D         64
#define WAVES     8
#define AGG_STRIDE 68    // 16x64 agg tile, padded row stride -> conflict-free b64 frag loads
#define WP_STRIDE  160   // K-pair group stride (128 data + 32 pad) -> conflict-free b64 B loads

__launch_bounds__(256, 1)
__global__ void gin_fused_kernel(const float* __restrict__ X,
                                 const float* __restrict__ W,
                                 const int*   __restrict__ rowptr,
                                 const int*   __restrict__ colidx,
                                 float*       __restrict__ out) {
    // weights, K-pair interleaved: lds_wp[(k>>1)*WP_STRIDE + c*2 + (k&1)] = W[k][c]
    __shared__ float lds_wp[32 * WP_STRIDE];              // 20480 B
    __shared__ float lds_agg[WAVES * 16 * AGG_STRIDE];    // 34816 B

    const int tid  = threadIdx.x;
    const int lane = tid & 31;
    const int wave = tid >> 5;

    // ---- stage weights (64x64 f32) into LDS, K-pair interleaved ----
    #pragma unroll
    for (int i = tid; i < D * D; i += 256) {
        const int k = i >> 6;
        const int c = i & 63;
        lds_wp[(k >> 1) * WP_STRIDE + c * 2 + (k & 1)] = W[i];
    }
    __syncthreads();

    // one wave handles a 16-node tile; force tile id scalar so edge-index
    // loads are provably wave-uniform (-> s_load)
    const int tile      = __builtin_amdgcn_readfirstlane((int)(blockIdx.x * WAVES + wave));
    const int node_base = tile * 16;

    float* aggT  = &lds_agg[wave * (16 * AGG_STRIDE)];
    const int c0 = lane * 2;   // each lane owns 2 consecutive feature columns

    // ---- SpMM: gather-sum 16 neighbors per node, write 16x64 tile to LDS ----
    for (int r = 0; r < 16; ++r) {
        int row = node_base + r;
        if (row >= N_NODES) row = N_NODES - 1;   // clamp: keep EXEC full for WMMA later
        const int  ebase = rowptr[row];
        const int* eb    = colidx + ebase;
        float s0 = 0.0f, s1 = 0.0f;
        #pragma unroll
        for (int e = 0; e < DEGREE; ++e) {
            const int col = eb[e];                         // wave-uniform address
            const v2f x2  = *(const v2f*)(X + (size_t)col * D + c0);
            s0 += x2.x;
            s1 += x2.y;
        }
        *(v2f*)(aggT + r * AGG_STRIDE + c0) = (v2f){s0, s1};
    }

    // fence the in-wave LDS transpose (store -> load, cross-lane)
    asm volatile("s_wait_dscnt 0x0" ::: "memory");

    // ---- A fragments: 16x4 f32 layout: lanes 0-15 K=k0..k0+1, lanes 16-31 K=k0+2..k0+3
    const int m    = lane & 15;
    const int koff = (lane >> 4) * 2;
    v2f a[16];
    #pragma unroll
    for (int ks = 0; ks < 16; ++ks) {
        a[ks] = *(const v2f*)(aggT + m * AGG_STRIDE + ks * 4 + koff);
    }

    // ---- GEMM: 16 K-steps x 4 N-tiles of v_wmma_f32_16x16x4_f32 ----
    const int n = lane & 15;
    v8f acc[4] = {};
    #pragma unroll
    for (int ks = 0; ks < 16; ++ks) {
        const int g = ks * 2 + (koff >> 1);   // K-pair group for this lane half
        #pragma unroll
        for (int nt = 0; nt < 4; ++nt) {
            // one b64: both K-pair values adjacent -> feeds WMMA without v_movs
            const v2f b = *(const v2f*)(lds_wp + g * WP_STRIDE + (nt * 16 + n) * 2);
            acc[nt] = __builtin_amdgcn_wmma_f32_16x16x4_f32(
                /*neg_a=*/false, a[ks], /*neg_b=*/false, b,
                /*c_mod=*/(short)0, acc[nt],
                /*reuse_a=*/false, /*reuse_b=*/false);
        }
    }

    // ---- store D: VGPR v -> M = v (lanes 0-15) / v+8 (lanes 16-31), N = lane&15
    // nontemporal: output is write-once; keep L2 free for the X gather set
    const int mb = (lane >> 4) * 8;
    #pragma unroll
    for (int nt = 0; nt < 4; ++nt) {
        #pragma unroll
        for (int v = 0; v < 8; ++v) {
            const int row = node_base + mb + v;
            if (row < N_NODES) {
                __builtin_nontemporal_store(acc[nt][v],
                                            out + (size_t)row * D + nt * 16 + n);
            }
        }
    }
}

extern "C" void kernel_launch(void* const* d_in, const int* in_sizes, int n_in,
                              void* d_out, int out_size, void* d_ws, size_t ws_size,
                              hipStream_t stream) {
    const float* X      = (const float*)d_in[0];
    const float* W      = (const float*)d_in[1];
    const int*   rowptr = (const int*)d_in[2];
    const int*   colidx = (const int*)d_in[3];
    float*       out    = (float*)d_out;

    const int tiles  = (N_NODES + 15) / 16;           // 6250
    const int blocks = (tiles + WAVES - 1) / WAVES;   // 782
    gin_fused_kernel<<<blocks, 256, 0, stream>>>(X, W, rowptr, colidx, out);
}